// MultiHeadAttention_85761906966848
// MI455X (gfx1250) — compile-verified
//
#include <hip/hip_runtime.h>
#include <cfloat>

// ---------------------------------------------------------------------------
// MI455X (gfx1250) implementation of "diagonal attention" MHA.
//   B=8, S=1024, D=1024, H=16, dk=64, fp32 throughout.
// All matrix math runs on the CDNA5 matrix pipe via V_WMMA_F32_16X16X4_F32.
// fp32 WMMA chosen over bf16: the whole working set (~176 MB) is L2-resident
// (192 MB L2, 23.3 TB/s HBM floor ~3.4 us), so the f32 matrix rate is not the
// bottleneck and fp32 keeps full numerical fidelity to the fp32 reference.
// Since K=4 per f32 WMMA, the LDS pipe is the binding constraint in the GEMM
// inner loop -> each wave computes a 32x64 tile so every B fragment feeds
// two row-groups (8 WMMAs per 6 LDS reads).
// ---------------------------------------------------------------------------

typedef __attribute__((ext_vector_type(2))) float v2f;
typedef __attribute__((ext_vector_type(8))) float v8f;

#define WMMA_F32_16x16x4(a, b, c) \
    __builtin_amdgcn_wmma_f32_16x16x4_f32(false, (a), false, (b), (short)0, (c), false, false)

// Problem constants
#define BATCH 8
#define SEQ   1024
#define DIM   1024
#define HEADS 16
#define DK    64

// ---------------------------------------------------------------------------
// Kernel 1: Out[M,N] = X[M,K] @ W[K,N] + bias[N]   (fp32 WMMA, LDS-tiled)
// Workgroup: 4 waves (128 threads) -> 128(M) x 64(N) output block.
// Each wave: 32 rows x 64 cols = 2 row-groups x 4 col-tiles = 8 C fragments.
// K staged in 64-deep chunks; LDS row stride 68 floats -> conflict-free
// b64 A-fragment reads (word banks 4*row+c, all 64 banks distinct).
//
// f32 WMMA fragment layout (wave32), lane L, lo=L&15, hi=L>>4:
//   A 16x4 : M = lo, K = {4t+2*hi, 4t+2*hi+1}  (VGPR0, VGPR1)
//   B 4x16 : N = lo, K = {4t+2*hi, 4t+2*hi+1}
//   C 16x16: VGPR r -> M = r + 8*hi, N = lo
// ---------------------------------------------------------------------------
__global__ __launch_bounds__(128) void gemm_bias_f32(
    const float* __restrict__ X, const float* __restrict__ W,
    const float* __restrict__ bias, float* __restrict__ Out,
    int M, int N, int K)
{
    __shared__ float Xs[128 * 68];   // 128 rows x 64 K (padded)
    __shared__ float Ws[64 * 68];    // 64 K x 64 cols (padded)

    const int tid  = threadIdx.x;
    const int wave = tid >> 5;
    const int lid  = tid & 31;
    const int lo   = lid & 15;
    const int hi   = lid >> 4;

    const int rowBlock = blockIdx.y * 128;
    const int colBase  = blockIdx.x * 64;

    v8f acc[2][4] = {};

    for (int kc = 0; kc < K; kc += 64) {
        __syncthreads();
        // Stage X[rowBlock..+127][kc..+63]: 2048 float4 slots, coalesced.
        #pragma unroll
        for (int i = 0; i < 16; ++i) {
            int idx = tid + i * 128;
            int r   = idx >> 4;
            int c4  = (idx & 15) << 2;
            const float4* xsrc = (const float4*)(X + (size_t)(rowBlock + r) * K + kc + c4);
            *(float4*)(&Xs[r * 68 + c4]) = *xsrc;
        }
        // Stage W[kc..+63][colBase..+63]: 1024 float4 slots, coalesced.
        #pragma unroll
        for (int i = 0; i < 8; ++i) {
            int idx = tid + i * 128;
            int r   = idx >> 4;
            int c4  = (idx & 15) << 2;
            const float4* wsrc = (const float4*)(W + (size_t)(kc + r) * N + colBase + c4);
            *(float4*)(&Ws[r * 68 + c4]) = *wsrc;
        }
        __syncthreads();

        const int lr0 = (wave << 5) + lo;        // row-group 0 A-row
        const int lr1 = lr0 + 16;                // row-group 1 A-row
        #pragma unroll
        for (int t = 0; t < 16; ++t) {
            const int k0 = 4 * t + 2 * hi;
            v2f a0, a1;
            a0.x = Xs[lr0 * 68 + k0];
            a0.y = Xs[lr0 * 68 + k0 + 1];
            a1.x = Xs[lr1 * 68 + k0];
            a1.y = Xs[lr1 * 68 + k0 + 1];
            #pragma unroll
            for (int ct = 0; ct < 4; ++ct) {
                const int lc = ct * 16 + lo;
                v2f b;
                b.x = Ws[k0 * 68 + lc];
                b.y = Ws[(k0 + 1) * 68 + lc];
                acc[0][ct] = WMMA_F32_16x16x4(a0, b, acc[0][ct]);
                acc[1][ct] = WMMA_F32_16x16x4(a1, b, acc[1][ct]);
            }
        }
    }

    // Epilogue: bias + store. Row of C VGPR r (group rg) is
    // rowBlock + wave*32 + rg*16 + r + 8*hi.
    #pragma unroll
    for (int rg = 0; rg < 2; ++rg) {
        const int row0 = rowBlock + (wave << 5) + (rg << 4) + (hi << 3);
        #pragma unroll
        for (int ct = 0; ct < 4; ++ct) {
            const int col = colBase + ct * 16 + lo;
            const float bv = bias[col];
            #pragma unroll
            for (int r = 0; r < 8; ++r) {
                Out[(size_t)(row0 + r) * N + col] = acc[rg][ct][r] + bv;
            }
        }
    }
}

// ---------------------------------------------------------------------------
// Kernel 2: per-(b,h,s) diagonal softmax weight, then scale V.
// One wave per (b, h, 16-row tile). Flash-style lane-local online softmax:
// each lane tracks (runMax, runSum) for its own column subset of 8 rows; a
// single 16-lane shfl_xor merge at the end produces the row (max, denom).
// The diagonal score is captured straight out of the C fragment (N == M).
// Writes weighted *in place over Q*: the wave preloads its Q rows first and
// is the only wave that ever touches that [16 x 64] region.
// ---------------------------------------------------------------------------
__global__ __launch_bounds__(128) void attn_diag_scale(
    const float* __restrict__ Q, const float* __restrict__ Km,
    const float* __restrict__ V, const int* __restrict__ lengths,
    float* __restrict__ Wt)
{
    const int lid = threadIdx.x & 31;
    const int lo  = lid & 15;
    const int hi  = lid >> 4;

    const int g  = blockIdx.x * 4 + (threadIdx.x >> 5);   // global wave id
    const int it = g & 63;            // row tile (S/16 = 64)
    const int h  = (g >> 6) & 15;     // head
    const int b  = g >> 10;           // batch

    const int s0  = it * 16;
    const int len = lengths[b];

    // Preload A fragments for Q rows s0..s0+15 (head h): 16 x v2f = 32 VGPRs.
    const float* qbase = Q + ((size_t)(b * SEQ + s0 + lo)) * DIM + h * DK;
    v2f aq[16];
    #pragma unroll
    for (int t = 0; t < 16; ++t)
        aq[t] = *(const v2f*)(qbase + 4 * t + 2 * hi);

    float mrow[8], srow[8];
    #pragma unroll
    for (int r = 0; r < 8; ++r) { mrow[r] = -FLT_MAX; srow[r] = 0.0f; }
    float diag_raw = -FLT_MAX;   // stays -FLT_MAX if this row's diag is masked

    for (int j = 0; j <= it; ++j) {
        const int q0 = j * 16;
        if (q0 >= len) break;                      // wave-uniform: all later tiles masked

        // Score tile: C[M][N] = q_{s0+M} . k_{q0+N}, K-dim = 64 via 16 WMMAs.
        const float* kbase = Km + ((size_t)(b * SEQ + q0 + lo)) * DIM + h * DK;
        v8f acc = {};
        #pragma unroll
        for (int t = 0; t < 16; ++t) {
            v2f kb = *(const v2f*)(kbase + 4 * t + 2 * hi);
            acc = WMMA_F32_16x16x4(aq[t], kb, acc);
        }

        // Mask + lane-local online softmax over this lane's column (q0+lo).
        const int q = q0 + lo;
        #pragma unroll
        for (int r = 0; r < 8; ++r) {
            const int s = s0 + r + 8 * hi;
            const float val = acc[r];
            const bool valid = (q <= s) && (q < len);
            const float nm = fmaxf(mrow[r], val);
            const float e1 = __expf(mrow[r] - nm);
            const float e2 = __expf(val - nm);
            if (valid) { srow[r] = srow[r] * e1 + e2; mrow[r] = nm; }
            if (valid && q == s) diag_raw = val;   // exactly one lane per row
        }
    }

    // Merge (max, sum) across the 16 lanes of each half (rows r+8*hi).
    float d_mine = 0.0f;
    #pragma unroll
    for (int r = 0; r < 8; ++r) {
        float m = mrow[r], sum = srow[r];
        #pragma unroll
        for (int off = 8; off >= 1; off >>= 1) {
            const float mo = __shfl_xor(m, off, 32);   // xor of bits 0..3: stays in half
            const float so = __shfl_xor(sum, off, 32);
            const float nm = fmaxf(m, mo);
            sum = sum * __expf(m - nm) + so * __expf(mo - nm);
            m = nm;
        }
        // Diagonal-capturing lane for row (r + 8*hi) is lane with lo == r + 8*hi.
        if (lo == r + 8 * hi) d_mine = __expf(diag_raw - m) / sum;  // 0 if masked
    }

    // weighted[b, s0+row, h*64 + :] = d[row] * V[...]; lane L -> row L>>1, half (L&1).
    const int orow = lid >> 1;
    const int src  = (orow < 8) ? orow : (orow + 16);   // capture lanes: 0-7 and 24-31
    const float d  = __shfl(d_mine, src, 32);

    const size_t off = ((size_t)(b * SEQ + s0 + orow)) * DIM + h * DK + (lid & 1) * 32;
    const float4* vsrc = (const float4*)(V + off);
    float4* odst = (float4*)(Wt + off);
    #pragma unroll
    for (int i = 0; i < 8; ++i) {
        float4 vv = vsrc[i];
        vv.x *= d; vv.y *= d; vv.z *= d; vv.w *= d;
        odst[i] = vv;
    }
}

// ---------------------------------------------------------------------------
// Host-side launcher. Scratch layout (needs 96 MB):
//   d_ws + 0       : Q  [8,1024,1024] f32  (overwritten in place with weighted)
//   d_ws + 32 MB   : K  [8,1024,1024] f32
//   d_ws + 64 MB   : V  [8,1024,1024] f32
// ---------------------------------------------------------------------------
extern "C" void kernel_launch(void* const* d_in, const int* in_sizes, int n_in,
                              void* d_out, int out_size, void* d_ws, size_t ws_size,
                              hipStream_t stream) {
    (void)in_sizes; (void)n_in; (void)out_size; (void)ws_size;

    const float* batch   = (const float*)d_in[0];
    const int*   lengths = (const int*)d_in[1];
    const float* wq = (const float*)d_in[2];
    const float* bq = (const float*)d_in[3];
    const float* wk = (const float*)d_in[4];
    const float* bk = (const float*)d_in[5];
    const float* wv = (const float*)d_in[6];
    const float* bv = (const float*)d_in[7];
    const float* w0 = (const float*)d_in[8];
    const float* b0 = (const float*)d_in[9];
    float* out = (float*)d_out;

    const size_t mat = (size_t)BATCH * SEQ * DIM;   // 8 Mi floats = 32 MB
    float* Qb = (float*)d_ws;
    float* Kb = Qb + mat;
    float* Vb = Kb + mat;

    const int M = BATCH * SEQ;                       // 8192
    const dim3 gemmGrid(DIM / 64, M / 128);          // (16, 64)
    const dim3 gemmBlock(128);

    // Q/K/V projections
    gemm_bias_f32<<<gemmGrid, gemmBlock, 0, stream>>>(batch, wq, bq, Qb, M, DIM, DIM);
    gemm_bias_f32<<<gemmGrid, gemmBlock, 0, stream>>>(batch, wk, bk, Kb, M, DIM, DIM);
    gemm_bias_f32<<<gemmGrid, gemmBlock, 0, stream>>>(batch, wv, bv, Vb, M, DIM, DIM);

    // Diagonal softmax weight + V scaling; writes weighted over Qb in place.
    const int totalWaves = BATCH * HEADS * (SEQ / 16);          // 8192
    attn_diag_scale<<<totalWaves / 4, 128, 0, stream>>>(Qb, Kb, Vb, lengths, Qb);

    // Output projection
    gemm_bias_f32<<<gemmGrid, gemmBlock, 0, stream>>>(Qb, w0, b0, out, M, DIM, DIM);
}